// KFAttention_1116691497368
// MI455X (gfx1250) — compile-verified
//
#include <hip/hip_runtime.h>
#include <hip/hip_bf16.h>
#include <math.h>

// ---------------------------------------------------------------------------
// Hyperbolic (Karcher-flow) attention for MI455X / gfx1250.
// All dense contractions via V_WMMA_F32_16X16X4_F32 (fp32 WMMA, wave32).
// NT GEMM stages B tiles into LDS with GLOBAL_LOAD_ASYNC_TO_LDS_B128
// (ASYNCcnt / s_wait_asynccnt), 8 waves share each staged 64x64 B chunk.
// Rank-1 Minkowski time-term folded into GEMM epilogues; logmap algebra
// factored so the Karcher inner loop is 2 GEMMs + 1 elementwise pass.
// Layout convention: embedded points stored time-LAST:
//   Q/Kh: [1024 x 272]  cols 0..255 spatial, col 256 time, pad 257..271 = 0
//   V/z : [1024 x 144]  cols 0..127 spatial, col 128 time, pad 129..143 = 0
// ---------------------------------------------------------------------------

typedef __attribute__((ext_vector_type(2))) float v2f;
typedef __attribute__((ext_vector_type(8))) float v8f;
typedef int v4i_b128 __attribute__((vector_size(16)));   // builtin's pointee type

#define EPS_F      1e-7f
#define TAN_CLIP_F 2.8f
#define BETA_F     0.0625f   // 1/sqrt(256)
// k (curvature) == 1.0 -> sqrt(k) == 1; folded into the math below.

// ---------------- CDNA5 async global->LDS copy helpers ---------------------
__device__ __forceinline__ void async_b128_to_lds(const float* g, float* l) {
#if __has_builtin(__builtin_amdgcn_global_load_async_to_lds_b128)
  // signature (probe-confirmed): (v4i AS1*, v4i AS3*, imm offset, imm cpol)
  __builtin_amdgcn_global_load_async_to_lds_b128(
      (__attribute__((address_space(1))) v4i_b128*)(uintptr_t)g,
      (__attribute__((address_space(3))) v4i_b128*)(unsigned int)(uintptr_t)l,
      0, 0);
#else
  const unsigned int loff = (unsigned int)(uintptr_t)l;  // LDS byte offset
  asm volatile("global_load_async_to_lds_b128 %0, %1, off"
               :: "v"(loff), "v"(g) : "memory");
#endif
}

__device__ __forceinline__ void wait_async0() {
#if __has_builtin(__builtin_amdgcn_s_wait_asynccnt)
  __builtin_amdgcn_s_wait_asynccnt(0);
#else
  asm volatile("s_wait_asynccnt 0x0" ::: "memory");
#endif
}

// ---------------- block reductions (blockDim.x == 256 everywhere) ----------
__device__ __forceinline__ float blockReduceSum(float v, float* sbuf) {
  int t = threadIdx.x;
  sbuf[t] = v; __syncthreads();
  for (int s = 128; s > 0; s >>= 1) {
    if (t < s) sbuf[t] += sbuf[t + s];
    __syncthreads();
  }
  float r = sbuf[0];
  __syncthreads();
  return r;
}

__device__ __forceinline__ float blockReduceMax(float v, float* sbuf) {
  int t = threadIdx.x;
  sbuf[t] = v; __syncthreads();
  for (int s = 128; s > 0; s >>= 1) {
    if (t < s) sbuf[t] = fmaxf(sbuf[t], sbuf[t + s]);
    __syncthreads();
  }
  float r = sbuf[0];
  __syncthreads();
  return r;
}

// ---------------------------------------------------------------------------
// NT GEMM: C[M x N] = A[M x K] @ B[N x K]^T  (+ coef * at(x)bt rank-1)
// Block = 8 waves -> 128 M-rows x 64 N-cols. B staged to LDS in 64x64 K
// chunks via async-to-LDS (each thread issues 4 b128s), stride-68 padding
// keeps the 16-lane float2 B-fragment reads bank-conflict free.
// Requires: M%128==0, N%64==0, K%64==0, 16B-aligned B rows.
// grid.x == (M/128)*(N/64), blockDim == 256.
// ---------------------------------------------------------------------------
__global__ __launch_bounds__(256) void gemm_nt_wmma(
    const float* __restrict__ A, const float* __restrict__ B,
    float* __restrict__ C, int M, int N, int K,
    int lda, int ldb, int ldc,
    const float* __restrict__ at, int ats,
    const float* __restrict__ bt, int bts, float coef)
{
  __shared__ float sB[64 * 68];               // 17.4 KB, stride 68 (16B rows)
  const int t    = threadIdx.x;
  const int lane = t & 31;
  const int wave = t >> 5;
  const int nGroups = N >> 6;                 // N / 64
  const int bM = blockIdx.x / nGroups;
  const int bN = blockIdx.x % nGroups;
  const int m0 = bM * 128 + wave * 16;
  const int n0 = bN * 64;
  const int r  = lane & 15;                   // row-in-tile (A) / col-in-tile (B/C)
  const int h  = lane >> 4;                   // half-wave: K offset 0 or 2

  const float* Ap = A + (size_t)(m0 + r) * lda + 2 * h;

  v8f acc0 = {}, acc1 = {}, acc2 = {}, acc3 = {};

  for (int kc = 0; kc < K; kc += 64) {
    // ---- stage B[n0:n0+64, kc:kc+64] into LDS (async, ASYNCcnt-tracked) ----
#pragma unroll
    for (int j = 0; j < 4; ++j) {
      const int f4  = j * 256 + t;            // float4 index within chunk
      const int row = f4 >> 4;                // 0..63
      const int cg  = f4 & 15;                // 0..15 (4-float groups)
      const float* g = B + (size_t)(n0 + row) * ldb + kc + cg * 4;
      async_b128_to_lds(g, &sB[row * 68 + cg * 4]);
    }
    __builtin_prefetch(Ap + kc + 64, 0, 1);   // gfx1250 global_prefetch_b8
    wait_async0();
    __syncthreads();

    // ---- 16 K-steps of 4 WMMAs, B fragments from LDS ----
#pragma unroll
    for (int kk = 0; kk < 64; kk += 4) {
      v2f a  = *(const v2f*)(Ap + kc + kk);   // A frag: K = kc+kk+2h .. +1
      const float* bb = &sB[kk + 2 * h];
      v2f b0 = *(const v2f*)(bb + (size_t)(r     ) * 68);
      v2f b1 = *(const v2f*)(bb + (size_t)(r + 16) * 68);
      v2f b2 = *(const v2f*)(bb + (size_t)(r + 32) * 68);
      v2f b3 = *(const v2f*)(bb + (size_t)(r + 48) * 68);
      acc0 = __builtin_amdgcn_wmma_f32_16x16x4_f32(false, a, false, b0, (short)0, acc0, false, false);
      acc1 = __builtin_amdgcn_wmma_f32_16x16x4_f32(false, a, false, b1, (short)0, acc1, false, false);
      acc2 = __builtin_amdgcn_wmma_f32_16x16x4_f32(false, a, false, b2, (short)0, acc2, false, false);
      acc3 = __builtin_amdgcn_wmma_f32_16x16x4_f32(false, a, false, b3, (short)0, acc3, false, false);
    }
    __syncthreads();                          // sB reused next chunk
  }

  // Rank-1 Minkowski time correction: C += coef * at[m] * bt[n]
  if (at != nullptr) {
    const float bt0 = bt[(size_t)(n0      + r) * bts];
    const float bt1 = bt[(size_t)(n0 + 16 + r) * bts];
    const float bt2 = bt[(size_t)(n0 + 32 + r) * bts];
    const float bt3 = bt[(size_t)(n0 + 48 + r) * bts];
#pragma unroll
    for (int v = 0; v < 8; ++v) {
      const float av = coef * at[(size_t)(m0 + v + 8 * h) * ats];
      acc0[v] += av * bt0;
      acc1[v] += av * bt1;
      acc2[v] += av * bt2;
      acc3[v] += av * bt3;
    }
  }

#pragma unroll
  for (int v = 0; v < 8; ++v) {
    const size_t row = (size_t)(m0 + v + 8 * h) * ldc;
    C[row + n0      + r] = acc0[v];
    C[row + n0 + 16 + r] = acc1[v];
    C[row + n0 + 32 + r] = acc2[v];
    C[row + n0 + 48 + r] = acc3[v];
  }
}

// ---------------------------------------------------------------------------
// NN GEMM: C[M x N] = A[M x K] @ B[K x N]. One 16x16 tile per wave.
// Requires M%16==0, N%16==0, K%4==0. grid.x * 8 waves == (M/16)*(N/16).
// ---------------------------------------------------------------------------
__global__ __launch_bounds__(256) void gemm_nn_wmma(
    const float* __restrict__ A, const float* __restrict__ B,
    float* __restrict__ C, int M, int N, int K,
    int lda, int ldb, int ldc)
{
  const int lane = threadIdx.x & 31;
  const int wave = threadIdx.x >> 5;
  const int gw   = blockIdx.x * 8 + wave;
  const int nTiles = N >> 4;
  const int mTile = gw / nTiles;
  const int nTile = gw % nTiles;
  if (mTile * 16 >= M) return;

  const int r  = lane & 15;
  const int h  = lane >> 4;
  const int m0 = mTile * 16;
  const int n0 = nTile * 16;

  const float* Ap = A + (size_t)(m0 + r) * lda + 2 * h;
  v8f acc = {};

  for (int kk = 0; kk < K; kk += 4) {
    v2f a = *(const v2f*)(Ap + kk);
    const int kb = kk + 2 * h;
    v2f b = { B[(size_t)kb * ldb + n0 + r],
              B[(size_t)(kb + 1) * ldb + n0 + r] };
    acc = __builtin_amdgcn_wmma_f32_16x16x4_f32(false, a, false, b, (short)0, acc, false, false);
  }

#pragma unroll
  for (int v = 0; v < 8; ++v)
    C[(size_t)(m0 + v + 8 * h) * ldc + n0 + r] = acc[v];
}

// ---------------------------------------------------------------------------
// expmap0(to_tangent0(X)) rowwise, time stored LAST (col D), pad zeroed.
// Optionally emits mink(P,P) per row (needed for V).
// ---------------------------------------------------------------------------
__global__ __launch_bounds__(256) void embed_expmap0(
    const float* __restrict__ X, int D,
    float* __restrict__ OUT, int ldo, float* __restrict__ minkOut)
{
  __shared__ float sbuf[256];
  const int s = blockIdx.x;
  const float* x = X + (size_t)s * D;
  float* o = OUT + (size_t)s * ldo;

  float ss = 0.f;
  for (int d = threadIdx.x; d < D; d += 256) { float v = x[d]; ss += v * v; }
  ss = blockReduceSum(ss, sbuf);

  const float nrm    = sqrtf(ss);
  const float factor = fminf(TAN_CLIP_F / (nrm + EPS_F), 1.0f);
  const float un     = nrm * factor;                 // theta (sqrt(k)=1)
  const float timev  = coshf(un);
  const float scale  = sinhf(un) / fmaxf(un, EPS_F) * factor;

  for (int d = threadIdx.x; d < D; d += 256) o[d] = scale * x[d];
  for (int d = D + 1 + threadIdx.x; d < ldo; d += 256) o[d] = 0.0f;
  if (threadIdx.x == 0) {
    o[D] = timev;
    if (minkOut) minkOut[s] = scale * scale * ss - timev * timev;
  }
}

// ---------------------------------------------------------------------------
// Rowwise softmax of BETA * S (in place). N = 1024.
// ---------------------------------------------------------------------------
__global__ __launch_bounds__(256) void softmax_rows(float* __restrict__ S, int N)
{
  __shared__ float sbuf[256];
  float* p = S + (size_t)blockIdx.x * N;
  float mx = -3.0e38f;
  for (int j = threadIdx.x; j < N; j += 256) mx = fmaxf(mx, BETA_F * p[j]);
  mx = blockReduceMax(mx, sbuf);
  float sum = 0.f;
  for (int j = threadIdx.x; j < N; j += 256) {
    float e = __expf(BETA_F * p[j] - mx);
    p[j] = e; sum += e;
  }
  sum = blockReduceSum(sum, sbuf);
  const float inv = 1.0f / sum;
  for (int j = threadIdx.x; j < N; j += 256) p[j] *= inv;
}

// z <- origin (time=1 at col 128), mzz <- mink(z,z) = -1
__global__ __launch_bounds__(256) void init_z(float* __restrict__ Z, float* __restrict__ MZZ)
{
  const int i = blockIdx.x * 256 + threadIdx.x;
  if (i < 1024 * 144) Z[i] = ((i % 144) == 128) ? 1.0f : 0.0f;
  if (i < 1024)       MZZ[i] = -1.0f;
}

// ---------------------------------------------------------------------------
// Per (s,m): dist = acosh(max(-ip, 1+eps));
// denom^2 = mink(V + ip*z, V + ip*z) = mvv[m] + ip^2*(2 + mzz[s])   (k=1)
// w = alpha * dist / denom  (overwrites IP);  G[s] = sum_m w*ip.
// ---------------------------------------------------------------------------
__global__ __launch_bounds__(256) void karcher_weights(
    float* __restrict__ IP, const float* __restrict__ AL,
    const float* __restrict__ MVV, const float* __restrict__ MZZ,
    float* __restrict__ G)
{
  __shared__ float sbuf[256];
  const int s = blockIdx.x;
  const float mz = MZZ[s];
  float* ipr = IP + (size_t)s * 1024;
  const float* ar = AL + (size_t)s * 1024;
  float g = 0.f;
  for (int m = threadIdx.x; m < 1024; m += 256) {
    const float ip    = ipr[m];
    const float dist  = acoshf(fmaxf(-ip, 1.0f + EPS_F));
    const float den2  = MVV[m] + ip * ip * (2.0f + mz);
    const float denom = sqrtf(fmaxf(den2, EPS_F));
    const float w     = ar[m] * dist / denom;
    ipr[m] = w;
    g += w * ip;
  }
  g = blockReduceSum(g, sbuf);
  if (threadIdx.x == 0) G[s] = g;
}

// ---------------------------------------------------------------------------
// z <- expmap(z, u) with u = EU_row + G[s]*z_row (129 dims, time at 128).
// Guarded by runtime karcher_steps (device scalar) for graph-capture safety.
// ---------------------------------------------------------------------------
__global__ __launch_bounds__(256) void karcher_update(
    const float* __restrict__ EU, const float* __restrict__ G,
    float* __restrict__ Z, float* __restrict__ MZZ,
    const int* __restrict__ nsteps, int step)
{
  if (step >= *nsteps) return;
  __shared__ float sbuf[256];
  __shared__ float s_ut, s_zt;
  const int s = blockIdx.x;
  const int d = threadIdx.x;
  const float Gs = G[s];
  const float zd = (d < 144) ? Z[(size_t)s * 144 + d] : 0.0f;
  const float ed = (d < 144) ? EU[(size_t)s * 144 + d] : 0.0f;
  const float u  = ed + Gs * zd;
  if (d == 128) s_ut = u;
  const float ssum = blockReduceSum((d < 128) ? u * u : 0.0f, sbuf);
  const float mink = ssum - s_ut * s_ut;
  const float nu = sqrtf(fmaxf(mink, EPS_F));
  const float ch = coshf(nu);
  const float sh = sinhf(nu) / nu;
  const float zn = ch * zd + sh * u;
  if (d < 144) Z[(size_t)s * 144 + d] = zn;
  if (d == 128) s_zt = zn;
  const float msum = blockReduceSum((d < 128) ? zn * zn : 0.0f, sbuf);
  if (d == 0) MZZ[s] = msum - s_zt * s_zt;
}

// out[s, 0:128] = logmap0(z)[1:] = acosh(max(z_time,1+eps)) * zs / max(|zs|,eps)
__global__ __launch_bounds__(256) void logmap0_out(
    const float* __restrict__ Z, float* __restrict__ OUT)
{
  __shared__ float sbuf[256];
  const int s = blockIdx.x;
  const int d = threadIdx.x;
  const float* zr = Z + (size_t)s * 144;
  const float zd = (d < 128) ? zr[d] : 0.0f;
  const float n2 = blockReduceSum(zd * zd, sbuf);
  const float n  = fmaxf(sqrtf(n2), EPS_F);
  const float theta = acoshf(fmaxf(zr[128], 1.0f + EPS_F));
  if (d < 128) OUT[(size_t)s * 128 + d] = theta * zd / n;
}

// ---------------------------------------------------------------------------
extern "C" void kernel_launch(void* const* d_in, const int* in_sizes, int n_in,
                              void* d_out, int out_size, void* d_ws, size_t ws_size,
                              hipStream_t stream) {
  const float* R  = (const float*)d_in[0];   // 1024 x 512
  const float* Y  = (const float*)d_in[1];   // 1024 x 512
  const float* WQ = (const float*)d_in[2];   // 256 x 512
  const float* WK = (const float*)d_in[3];   // 256 x 512
  const float* WV = (const float*)d_in[4];   // 128 x 256
  const int* nsteps = (const int*)d_in[5];   // karcher_steps (device scalar)
  float* out = (float*)d_out;                // 1024 x 128
  float* ws  = (float*)d_ws;

  // workspace carve-up (floats, all 64B aligned) ~15.0 MB
  float* XQ  = ws;                    // 1024*256
  float* KY  = XQ  + 262144;          // 1024*256
  float* XV  = KY  + 262144;          // 1024*128
  float* Qf  = XV  + 131072;          // 1024*272
  float* Kf  = Qf  + 278528;          // 1024*272
  float* Vf  = Kf  + 278528;          // 1024*144
  float* AL  = Vf  + 147456;          // 1024*1024 (sims -> alpha)
  float* IPW = AL  + 1048576;         // 1024*1024 (ip -> weights)
  float* EU  = IPW + 1048576;         // 1024*144
  float* Z   = EU  + 147456;          // 1024*144
  float* G   = Z   + 147456;          // 1024
  float* MZZ = G   + 1024;            // 1024
  float* MVV = MZZ + 1024;            // 1024

  dim3 blk(256);

  // Projections (NT GEMMs: fp32 WMMA + async-to-LDS B staging)
  gemm_nt_wmma<<< 32, blk, 0, stream>>>(R,  WQ, XQ, 1024, 256, 512, 512, 512, 256,
                                        nullptr, 0, nullptr, 0, 0.f);
  gemm_nt_wmma<<< 32, blk, 0, stream>>>(Y,  WK, KY, 1024, 256, 512, 512, 512, 256,
                                        nullptr, 0, nullptr, 0, 0.f);
  gemm_nt_wmma<<< 16, blk, 0, stream>>>(KY, WV, XV, 1024, 128, 256, 256, 256, 128,
                                        nullptr, 0, nullptr, 0, 0.f);

  // Lift to hyperboloid
  embed_expmap0<<<1024, blk, 0, stream>>>(XQ, 256, Qf, 272, nullptr);
  embed_expmap0<<<1024, blk, 0, stream>>>(KY, 256, Kf, 272, nullptr);
  embed_expmap0<<<1024, blk, 0, stream>>>(XV, 128, Vf, 144, MVV);

  // sims = Qs@Ks^T - Qt(x)Kt  (rank-1 folded into epilogue), then softmax
  gemm_nt_wmma<<<128, blk, 0, stream>>>(Qf, Kf, AL, 1024, 1024, 256, 272, 272, 1024,
                                        Qf + 256, 272, Kf + 256, 272, -1.0f);
  softmax_rows<<<1024, blk, 0, stream>>>(AL, 1024);

  init_z<<<576, blk, 0, stream>>>(Z, MZZ);

  // Fixed 4 launches; z-update guarded by runtime *nsteps (correct for <=4).
  for (int step = 0; step < 4; ++step) {
    gemm_nt_wmma<<<128, blk, 0, stream>>>(Z, Vf, IPW, 1024, 1024, 128, 144, 144, 1024,
                                          Z + 128, 144, Vf + 128, 144, -1.0f);
    karcher_weights<<<1024, blk, 0, stream>>>(IPW, AL, MVV, MZZ, G);
    gemm_nn_wmma<<< 72, blk, 0, stream>>>(IPW, Vf, EU, 1024, 144, 1024, 1024, 144, 144);
    karcher_update<<<1024, blk, 0, stream>>>(EU, G, Z, MZZ, nsteps, step);
  }

  logmap0_out<<<1024, blk, 0, stream>>>(Z, out);
}